// Equalize_62182536512372
// MI455X (gfx1250) — compile-verified
//
#include <hip/hip_runtime.h>

// ---------------------------------------------------------------------------
// Equalize (AutoAugment) for x: [64, 512, 512, 3] fp32 in [0,1].
// Memory-bound: ~604 MB min traffic -> ~26 us at 23.3 TB/s. No WMMA applies;
// we lean on gfx1250 paths instead: async global->LDS staging (ASYNCcnt),
// LDS ds_add atomics for per-block histograms, global_prefetch_b8 streaming.
// ---------------------------------------------------------------------------

#define NBINS   256
#define BATCHES 64
#define HW      (512 * 512)
#define CHANNELS 3
#define PER_BATCH (HW * CHANNELS)          // 786432 floats per batch

#define THREADS 256                         // 8 wave32 waves per block
#define HIST_BLOCKS_PER_BATCH 32            // 24576 floats per block
#define HIST_TILE (THREADS * 4)             // 1024 floats per tile (b128/lane)
#define HIST_TILES (PER_BATCH / HIST_BLOCKS_PER_BATCH / HIST_TILE)  // 24
#define MAP_BLOCKS_PER_BATCH 96             // 8192 floats per block

#define AS1 __attribute__((address_space(1)))
#define AS3 __attribute__((address_space(3)))

typedef int v4i_ __attribute__((vector_size(16)));   // int4 vector for builtin

__device__ __forceinline__ int quant255(float v) {
  // clip(round(x*255), 0, 255) with round-to-nearest-even (matches jnp.round)
  float f = fminf(fmaxf(v * 255.0f, 0.0f), 255.0f);
  return (int)rintf(f);
}

// --- async global -> LDS copy of 16 bytes per lane (gfx1250 path) ----------
__device__ __forceinline__ void async_copy_b128(const float* gsrc, float* ldst) {
#if __has_builtin(__builtin_amdgcn_global_load_async_to_lds_b128)
  __builtin_amdgcn_global_load_async_to_lds_b128(
      (AS1 v4i_*)gsrc, (AS3 v4i_*)ldst, /*offset=*/0, /*cpol=*/0);
#else
  // Fallback: synchronous VGPR round-trip (still correct).
  float4 v = *(const float4*)gsrc;
  *(float4*)ldst = v;
#endif
}

template <int N>
__device__ __forceinline__ void wait_asynccnt() {
#if __has_builtin(__builtin_amdgcn_s_wait_asynccnt)
  __builtin_amdgcn_s_wait_asynccnt(N);
#else
  if constexpr (N == 0) asm volatile("s_wait_asynccnt 0" ::: "memory");
  else                  asm volatile("s_wait_asynccnt 1" ::: "memory");
#endif
}

// ---------------------------------------------------------------------------
// Pass 0: zero the 192 global histograms (in d_ws).
// ---------------------------------------------------------------------------
__global__ void eq_zero_hist(unsigned int* __restrict__ ghist) {
  ghist[blockIdx.x * blockDim.x + threadIdx.x] = 0u;
}

// ---------------------------------------------------------------------------
// Pass 1: per-(batch,channel) histograms. Double-buffered async staging of the
// pixel stream into LDS, LDS-private 3x256 histogram via ds_add_u32, flushed
// with global_atomic_add_u32.
// ---------------------------------------------------------------------------
__global__ void eq_hist(const float* __restrict__ x,
                        unsigned int* __restrict__ ghist) {
  __shared__ unsigned int lh[CHANNELS * NBINS];    // 3 KB
  __shared__ float stage[2][HIST_TILE];            // 8 KB double buffer

  const int t   = threadIdx.x;
  const int b   = blockIdx.x / HIST_BLOCKS_PER_BATCH;
  const int blk = blockIdx.x % HIST_BLOCKS_PER_BATCH;
  const int perBlock = PER_BATCH / HIST_BLOCKS_PER_BATCH;
  const int base = b * PER_BATCH + blk * perBlock;

  for (int i = t; i < CHANNELS * NBINS; i += THREADS) lh[i] = 0u;
  __syncthreads();

  // Prologue: stage tile 0.
  async_copy_b128(x + base + t * 4, &stage[0][t * 4]);

  for (int tile = 0; tile < HIST_TILES; ++tile) {
    const int cur = tile & 1;
    if (tile + 1 < HIST_TILES) {
      async_copy_b128(x + base + (tile + 1) * HIST_TILE + t * 4,
                      &stage[cur ^ 1][t * 4]);
      wait_asynccnt<1>();   // tile just issued may stay in flight
    } else {
      wait_asynccnt<0>();
    }
    const int idxInBatch = blk * perBlock + tile * HIST_TILE + t * 4;
#pragma unroll
    for (int k = 0; k < 4; ++k) {
      const float v = stage[cur][t * 4 + k];
      const int q = quant255(v);
      const int c = (idxInBatch + k) % CHANNELS;   // NHWC: channel innermost
      atomicAdd(&lh[c * NBINS + q], 1u);           // ds_add_u32
    }
  }
  __syncthreads();

  unsigned int* gh = ghist + b * CHANNELS * NBINS;
  for (int i = t; i < CHANNELS * NBINS; i += THREADS) {
    const unsigned int v = lh[i];
    if (v) atomicAdd(&gh[i], v);                   // global_atomic_add_u32
  }
}

// ---------------------------------------------------------------------------
// Pass 2: one block per (batch,channel): scan -> step -> LUT (as fp32/255).
// ---------------------------------------------------------------------------
__global__ void eq_lut(const unsigned int* __restrict__ ghist,
                       float* __restrict__ lutf) {
  __shared__ int cum[NBINS];
  __shared__ int hh[NBINS];
  __shared__ int red[NBINS];

  const int t  = threadIdx.x;
  const int bc = blockIdx.x;

  const int h = (int)ghist[bc * NBINS + t];
  hh[t]  = h;
  cum[t] = h;
  red[t] = (h > 0) ? t : -1;
  __syncthreads();

  // Hillis-Steele inclusive scan over 256 bins.
#pragma unroll
  for (int off = 1; off < NBINS; off <<= 1) {
    const int add = (t >= off) ? cum[t - off] : 0;
    __syncthreads();
    cum[t] += add;
    __syncthreads();
  }
  // Last nonzero bin (max-reduce of indices).
#pragma unroll
  for (int off = NBINS / 2; off > 0; off >>= 1) {
    if (t < off) red[t] = max(red[t], red[t + off]);
    __syncthreads();
  }
  const int last_idx = red[0];
  const int last_val = hh[last_idx];
  const int total    = cum[NBINS - 1];
  const int step     = (total - last_val) / (NBINS - 1);
  const int safe     = (step > 0) ? step : 1;

  int lut;
  if (step == 0) {
    lut = t;                                   // identity when histogram flat
  } else {
    const int prev = (t == 0) ? 0 : cum[t - 1];   // shifted cumsum, lut[0]=0
    lut = (prev + safe / 2) / safe;
    lut = min(max(lut, 0), NBINS - 1);
  }
  lutf[bc * NBINS + t] = (float)lut * (1.0f / 255.0f);
}

// ---------------------------------------------------------------------------
// Pass 3: gather every pixel through its (batch,channel) LUT. LUT cached in
// LDS (768 floats/batch); float4 streaming with global_prefetch_b8.
// ---------------------------------------------------------------------------
__global__ void eq_map(const float* __restrict__ x,
                       const float* __restrict__ lutf,
                       float* __restrict__ out) {
  __shared__ float llut[CHANNELS * NBINS];

  const int t   = threadIdx.x;
  const int b   = blockIdx.x / MAP_BLOCKS_PER_BATCH;
  const int blk = blockIdx.x % MAP_BLOCKS_PER_BATCH;

  for (int i = t; i < CHANNELS * NBINS; i += THREADS)
    llut[i] = lutf[b * CHANNELS * NBINS + i];
  __syncthreads();

  const int perBlock = PER_BATCH / MAP_BLOCKS_PER_BATCH;   // 8192
  const int base = b * PER_BATCH + blk * perBlock;
  const int nvec = perBlock / 4;                           // 2048 float4

  for (int i = t; i < nvec; i += THREADS) {
    const int off = base + i * 4;
    __builtin_prefetch(x + off + 4 * THREADS, 0, 0);       // global_prefetch_b8
    const float4 v = *(const float4*)(x + off);
    const int inB = blk * perBlock + i * 4;
    const int c0 = inB % CHANNELS;
    const int c1 = (c0 + 1) % CHANNELS;
    const int c2 = (c1 + 1) % CHANNELS;
    float4 o;
    o.x = llut[c0 * NBINS + quant255(v.x)];
    o.y = llut[c1 * NBINS + quant255(v.y)];
    o.z = llut[c2 * NBINS + quant255(v.z)];
    o.w = llut[c0 * NBINS + quant255(v.w)];
    *(float4*)(out + off) = o;
  }
}

// ---------------------------------------------------------------------------
extern "C" void kernel_launch(void* const* d_in, const int* in_sizes, int n_in,
                              void* d_out, int out_size, void* d_ws, size_t ws_size,
                              hipStream_t stream) {
  const float* x = (const float*)d_in[0];   // [64,512,512,3] fp32
  // d_in[1] = magnitude (int, unused by the reference)
  float* out = (float*)d_out;

  unsigned int* ghist = (unsigned int*)d_ws;                       // 192*256 u32
  float* lutf = (float*)((char*)d_ws + BATCHES * CHANNELS * NBINS * sizeof(unsigned int));

  eq_zero_hist<<<BATCHES * CHANNELS, NBINS, 0, stream>>>(ghist);
  eq_hist<<<BATCHES * HIST_BLOCKS_PER_BATCH, THREADS, 0, stream>>>(x, ghist);
  eq_lut<<<BATCHES * CHANNELS, NBINS, 0, stream>>>(ghist, lutf);
  eq_map<<<BATCHES * MAP_BLOCKS_PER_BATCH, THREADS, 0, stream>>>(x, lutf, out);

  (void)in_sizes; (void)n_in; (void)out_size; (void)ws_size;
}